// GraphQNetwork_52338471469345
// MI455X (gfx1250) — compile-verified
//
#include <hip/hip_runtime.h>

// ---- static sizes (match reference) ----
#define NN   524288      // nodes
#define EE   6291456     // edges
#define ET   (EE + NN)   // edges + self loops
#define BB   512         // graphs
#define NPG  1024        // nodes per graph
#define CH   18          // concat channels
#define HEADS 6
#define KS   8           // PMA seeds
#define EPSN 1e-5f

typedef __attribute__((ext_vector_type(2))) float v2f;
typedef __attribute__((ext_vector_type(8))) float v8f;

// ---------- order-preserving float <-> uint for atomicMax ----------
__device__ __forceinline__ unsigned enc_f(float s) {
    unsigned u = __float_as_uint(s);
    return (u & 0x80000000u) ? ~u : (u | 0x80000000u);
}
__device__ __forceinline__ float dec_f(unsigned v) {
    unsigned u = (v & 0x80000000u) ? (v & 0x7fffffffu) : ~v;
    return __uint_as_float(u);
}

// =====================================================================
// Dense projection Y[M,NOUT] = X[M,KD](ldX=LDX) @ W[KD,NOUT] + bias.
// f32 WMMA (V_WMMA_F32_16X16X4_F32), compile-time dims, LDS-staged tiles:
//   - block = 8 waves, 128 rows per block (M must be a multiple of 128)
//   - X tile / W / bias zero-padded in LDS -> no predication in compute
//   - fully unrolled WMMA chain per n-tile; EXEC all-ones throughout
// A layout: lanes 0-15 hold K=kc+{0,1}, lanes 16-31 hold K=kc+{2,3}.
// B layout: row-striped across lanes (lane&15 = column), same K split.
// D layout: VGPR i -> row (i + 8*(lane>>4)), col = lane&15.
// =====================================================================
template<int KD, int NOUT, int LDX>
__global__ __launch_bounds__(256) void gemm_wmma(const float* __restrict__ X,
                                                 const float* __restrict__ W,
                                                 const float* __restrict__ bias,
                                                 float* __restrict__ Y) {
    constexpr int KPAD = (KD + 3) & ~3;
    constexpr int NT   = (NOUT + 15) / 16;
    constexpr int NPAD = NT * 16;
    __shared__ float sX[128 * KPAD];
    __shared__ float sW[KPAD * NPAD];
    __shared__ float sB[NPAD];
    int tid = threadIdx.x;
    int rowBlk = blockIdx.x * 128;
    // weights + bias into LDS (zero padded)
    for (int i = tid; i < KPAD * NPAD; i += 256) {
        int kk = i / NPAD, nn = i % NPAD;
        sW[i] = (kk < KD && nn < NOUT) ? W[kk * NOUT + nn] : 0.f;
    }
    if (tid < NPAD) sB[tid] = (tid < NOUT) ? bias[tid] : 0.f;
    // X tile into LDS (zero padded in K)
    for (int i = tid; i < 128 * KPAD; i += 256) {
        int rr = i / KPAD, cc = i % KPAD;
        sX[i] = (cc < KD) ? X[(size_t)(rowBlk + rr) * LDX + cc] : 0.f;
    }
    __syncthreads();
    int lane = tid & 31, wave = tid >> 5;
    int hi = lane >> 4, r = lane & 15;
    const float* xw = &sX[(wave * 16 + r) * KPAD];
#pragma unroll
    for (int nt = 0; nt < NT; ++nt) {
        v8f acc = {};
#pragma unroll
        for (int kc = 0; kc < KPAD; kc += 4) {
            v2f a, b;
            a.x = xw[kc + hi * 2];
            a.y = xw[kc + hi * 2 + 1];
            b.x = sW[(kc + hi * 2)     * NPAD + nt * 16 + r];
            b.y = sW[(kc + hi * 2 + 1) * NPAD + nt * 16 + r];
            acc = __builtin_amdgcn_wmma_f32_16x16x4_f32(
                false, a, false, b, (short)0, acc, false, false);
        }
        int col = nt * 16 + r;
        if (col < NOUT) {
            float bv = sB[col];
#pragma unroll
            for (int i = 0; i < 8; ++i) {
                int row = rowBlk + wave * 16 + i + hi * 8;
                Y[(size_t)row * NOUT + col] = acc[i] + bv;
            }
        }
    }
}

// =====================================================================
// degree + self-loop edge_attr mean (PyG fill_value='mean')
// =====================================================================
__global__ void deg_loop_accum(const int* __restrict__ dst,
                               const float* __restrict__ ea,
                               float* __restrict__ deg,
                               float* __restrict__ lattr) {
    int e = blockIdx.x * blockDim.x + threadIdx.x;
    if (e >= EE) return;
    int d = dst[e];
    atomicAdd(&deg[d], 1.f);
#pragma unroll
    for (int j = 0; j < 4; ++j) atomicAdd(&lattr[(size_t)d * 4 + j], ea[(size_t)e * 4 + j]);
}

__global__ void deg_loop_final(float* __restrict__ lattr, const float* __restrict__ deg) {
    int n = blockIdx.x * blockDim.x + threadIdx.x;
    if (n >= NN) return;
    float dd = fmaxf(deg[n], 1.f);
#pragma unroll
    for (int j = 0; j < 4; ++j) lattr[(size_t)n * 4 + j] /= dd;
}

// =====================================================================
// GATv2 edge passes (edges + self-loops concatenated at e >= EE)
// =====================================================================
__global__ void edge_score(const float* __restrict__ xl, const float* __restrict__ xr,
                           const int* __restrict__ src, const int* __restrict__ dst,
                           const float* __restrict__ ea, const float* __restrict__ lattr,
                           const float* __restrict__ We, const float* __restrict__ att,
                           int fo, float* __restrict__ sbuf, unsigned* __restrict__ smax) {
    int e = blockIdx.x * blockDim.x + threadIdx.x;
    if (e >= ET) return;
    int s_, d_;
    float eav[4];
    if (e < EE) {
        s_ = src[e]; d_ = dst[e];
#pragma unroll
        for (int j = 0; j < 4; ++j) eav[j] = ea[(size_t)e * 4 + j];
    } else {
        int n = e - EE; s_ = n; d_ = n;
#pragma unroll
        for (int j = 0; j < 4; ++j) eav[j] = lattr[(size_t)n * 4 + j];
    }
    float sv = 0.f;
    for (int c = 0; c < fo; ++c) {
        float m = xl[(size_t)s_ * fo + c] + xr[(size_t)d_ * fo + c];
#pragma unroll
        for (int j = 0; j < 4; ++j) m += eav[j] * We[j * fo + c];
        float l = m > 0.f ? m : 0.2f * m;       // LeakyReLU(0.2)
        sv += l * att[c];
    }
    sbuf[e] = sv;
    atomicMax(&smax[d_], enc_f(sv));
}

__global__ void edge_exp(const int* __restrict__ dst, float* __restrict__ sbuf,
                         const unsigned* __restrict__ smax, float* __restrict__ denom) {
    int e = blockIdx.x * blockDim.x + threadIdx.x;
    if (e >= ET) return;
    int d_ = (e < EE) ? dst[e] : (e - EE);
    float ex = __expf(sbuf[e] - dec_f(smax[d_]));
    sbuf[e] = ex;
    atomicAdd(&denom[d_], ex);
}

__global__ void edge_scatter(const float* __restrict__ xl,
                             const int* __restrict__ src, const int* __restrict__ dst,
                             const float* __restrict__ exbuf, const float* __restrict__ denom,
                             int fo, float* __restrict__ outp) {
    int e = blockIdx.x * blockDim.x + threadIdx.x;
    if (e >= ET) return;
    int s_, d_;
    if (e < EE) { s_ = src[e]; d_ = dst[e]; }
    else        { s_ = e - EE; d_ = s_; }
    float alpha = exbuf[e] / denom[d_];
    for (int c = 0; c < fo; ++c)
        atomicAdd(&outp[(size_t)d_ * fo + c], alpha * xl[(size_t)s_ * fo + c]);
}

// =====================================================================
// GraphNorm (+conv bias, +ReLU) fused; writes into strided xc concat
// one workgroup per graph (nodes of a graph are contiguous)
// =====================================================================
__global__ void graphnorm_relu(const float* __restrict__ y, const float* __restrict__ cb,
                               const float* __restrict__ w, const float* __restrict__ bn,
                               const float* __restrict__ ms, int fo,
                               float* __restrict__ xc, int colOff) {
    __shared__ float red[256];
    __shared__ float meanS[8], varS[8];
    int g = blockIdx.x, tid = threadIdx.x;
    const float* yg = y + (size_t)g * NPG * fo;
    float part[8];
    for (int c = 0; c < fo; ++c) part[c] = 0.f;
    for (int i = tid; i < NPG; i += 256)
        for (int c = 0; c < fo; ++c) part[c] += yg[(size_t)i * fo + c] + cb[c];
    for (int c = 0; c < fo; ++c) {
        red[tid] = part[c]; __syncthreads();
        for (int s = 128; s > 0; s >>= 1) { if (tid < s) red[tid] += red[tid + s]; __syncthreads(); }
        if (tid == 0) meanS[c] = red[0] * (1.f / NPG);
        __syncthreads();
    }
    for (int c = 0; c < fo; ++c) part[c] = 0.f;
    for (int i = tid; i < NPG; i += 256)
        for (int c = 0; c < fo; ++c) {
            float cc = yg[(size_t)i * fo + c] + cb[c] - ms[c] * meanS[c];
            part[c] += cc * cc;
        }
    for (int c = 0; c < fo; ++c) {
        red[tid] = part[c]; __syncthreads();
        for (int s = 128; s > 0; s >>= 1) { if (tid < s) red[tid] += red[tid + s]; __syncthreads(); }
        if (tid == 0) varS[c] = red[0] * (1.f / NPG);
        __syncthreads();
    }
    for (int i = tid; i < NPG; i += 256)
        for (int c = 0; c < fo; ++c) {
            float cc = yg[(size_t)i * fo + c] + cb[c] - ms[c] * meanS[c];
            float v = w[c] * cc * rsqrtf(varS[c] + EPSN) + bn[c];
            xc[((size_t)g * NPG + i) * CH + colOff + c] = fmaxf(v, 0.f);
        }
}

// =====================================================================
// small projection: qp[rows,CH] = S[rows,CH] @ Wq + bq
// =====================================================================
__global__ void small_proj(const float* __restrict__ S, const float* __restrict__ Wq,
                           const float* __restrict__ bq, float* __restrict__ qp, int rows) {
    int t = blockIdx.x * blockDim.x + threadIdx.x;
    if (t >= rows * CH) return;
    int i = t / CH, j = t % CH;
    float a = bq[j];
    for (int c = 0; c < CH; ++c) a += S[i * CH + c] * Wq[c * CH + j];
    qp[t] = a;
}

// =====================================================================
// PMA_8 attention: one block per graph, 8 waves, wave w owns query w.
// Two passes over the 1024 keys (max, then exp/sum + V accumulation),
// wave-level shfl_xor reductions; Wo residual fused in-block.
// =====================================================================
__global__ void pma1_attn(const float* __restrict__ Kp, const float* __restrict__ Vp,
                          const float* __restrict__ qp1,
                          const float* __restrict__ Wo, const float* __restrict__ bo,
                          float* __restrict__ hout) {
    __shared__ float sQ[KS * CH];
    __shared__ float sO[KS * CH];
    int g = blockIdx.x, tid = threadIdx.x;
    if (tid < KS * CH) sQ[tid] = qp1[tid];
    __syncthreads();
    int wave = tid >> 5, lane = tid & 31;
    int q = wave;                                   // 8 waves = 8 queries
    float qr[CH];
#pragma unroll
    for (int j = 0; j < CH; ++j) qr[j] = sQ[q * CH + j];
    const float scale = 0.2357022603955158f;        // 1/sqrt(18)
    const float* Kg = Kp + (size_t)g * NPG * CH;
    const float* Vg = Vp + (size_t)g * NPG * CH;
    float mx[HEADS];
#pragma unroll
    for (int h = 0; h < HEADS; ++h) mx[h] = -3.0e38f;
    for (int k = lane; k < NPG; k += 32) {
        const float* kr = Kg + (size_t)k * CH;
#pragma unroll
        for (int h = 0; h < HEADS; ++h) {
            float s = (qr[h*3]*kr[h*3] + qr[h*3+1]*kr[h*3+1] + qr[h*3+2]*kr[h*3+2]) * scale;
            mx[h] = fmaxf(mx[h], s);
        }
    }
#pragma unroll
    for (int h = 0; h < HEADS; ++h)
        for (int off = 16; off; off >>= 1) mx[h] = fmaxf(mx[h], __shfl_xor(mx[h], off, 32));
    float den[HEADS], acc[HEADS][3];
#pragma unroll
    for (int h = 0; h < HEADS; ++h) { den[h] = 0.f; acc[h][0]=acc[h][1]=acc[h][2]=0.f; }
    for (int k = lane; k < NPG; k += 32) {
        const float* kr = Kg + (size_t)k * CH;
        const float* vr = Vg + (size_t)k * CH;
#pragma unroll
        for (int h = 0; h < HEADS; ++h) {
            float s = (qr[h*3]*kr[h*3] + qr[h*3+1]*kr[h*3+1] + qr[h*3+2]*kr[h*3+2]) * scale;
            float e = __expf(s - mx[h]);
            den[h] += e;
            acc[h][0] += e * vr[h*3]; acc[h][1] += e * vr[h*3+1]; acc[h][2] += e * vr[h*3+2];
        }
    }
#pragma unroll
    for (int h = 0; h < HEADS; ++h)
        for (int off = 16; off; off >>= 1) {
            den[h]    += __shfl_xor(den[h],    off, 32);
            acc[h][0] += __shfl_xor(acc[h][0], off, 32);
            acc[h][1] += __shfl_xor(acc[h][1], off, 32);
            acc[h][2] += __shfl_xor(acc[h][2], off, 32);
        }
    if (lane == 0) {
#pragma unroll
        for (int h = 0; h < HEADS; ++h)
            for (int d = 0; d < 3; ++d)
                sO[q * CH + h*3 + d] = qr[h*3 + d] + acc[h][d] / den[h];
    }
    __syncthreads();
    if (tid < KS * CH) {
        int i = tid / CH, j = tid % CH;
        float t = bo[j];
        for (int c = 0; c < CH; ++c) t += sO[i * CH + c] * Wo[c * CH + j];
        hout[(size_t)g * KS * CH + tid] = sO[tid] + fmaxf(t, 0.f);
    }
}

// =====================================================================
// SAB: MAB(h, h), 8x8 attention per graph, fully in LDS
// =====================================================================
__global__ void sab_kernel(const float* __restrict__ hin,
                           const float* __restrict__ Wq, const float* __restrict__ bq,
                           const float* __restrict__ Wk, const float* __restrict__ bk,
                           const float* __restrict__ Wv, const float* __restrict__ bv,
                           const float* __restrict__ Wo, const float* __restrict__ bo,
                           float* __restrict__ hout) {
    __shared__ float X[KS*CH], Qp[KS*CH], Kp[KS*CH], Vp[KS*CH], O[KS*CH];
    int g = blockIdx.x, tid = threadIdx.x;
    if (tid < KS * CH) X[tid] = hin[(size_t)g * KS * CH + tid];
    __syncthreads();
    if (tid < KS * CH) {
        int i = tid / CH, j = tid % CH;
        float aq = bq[j], ak = bk[j], av = bv[j];
        for (int c = 0; c < CH; ++c) {
            float xv = X[i * CH + c];
            aq += xv * Wq[c * CH + j]; ak += xv * Wk[c * CH + j]; av += xv * Wv[c * CH + j];
        }
        Qp[tid] = aq; Kp[tid] = ak; Vp[tid] = av;
    }
    __syncthreads();
    if (tid < KS * HEADS) {
        int q = tid / HEADS, h = tid % HEADS;
        const float scale = 0.2357022603955158f;
        float sc[KS], m = -3.0e38f;
        for (int k = 0; k < KS; ++k) {
            float s = 0.f;
            for (int d = 0; d < 3; ++d) s += Qp[q*CH + h*3 + d] * Kp[k*CH + h*3 + d];
            sc[k] = s * scale; m = fmaxf(m, sc[k]);
        }
        float dsum = 0.f;
        for (int k = 0; k < KS; ++k) { sc[k] = __expf(sc[k] - m); dsum += sc[k]; }
        for (int d = 0; d < 3; ++d) {
            float o = 0.f;
            for (int k = 0; k < KS; ++k) o += sc[k] * Vp[k*CH + h*3 + d];
            O[q*CH + h*3 + d] = Qp[q*CH + h*3 + d] + o / dsum;
        }
    }
    __syncthreads();
    if (tid < KS * CH) {
        int i = tid / CH, j = tid % CH;
        float t = bo[j];
        for (int c = 0; c < CH; ++c) t += O[i*CH + c] * Wo[c*CH + j];
        hout[(size_t)g * KS * CH + tid] = O[tid] + fmaxf(t, 0.f);
    }
}

// =====================================================================
// PMA_1: MAB(S2, h), 1x8 attention per graph
// =====================================================================
__global__ void pma2_kernel(const float* __restrict__ hin, const float* __restrict__ S2,
                            const float* __restrict__ Wq, const float* __restrict__ bq,
                            const float* __restrict__ Wk, const float* __restrict__ bk,
                            const float* __restrict__ Wv, const float* __restrict__ bv,
                            const float* __restrict__ Wo, const float* __restrict__ bo,
                            float* __restrict__ pooled) {
    __shared__ float X[KS*CH], Qp[CH], Kp[KS*CH], Vp[KS*CH], O[CH];
    int g = blockIdx.x, tid = threadIdx.x;
    if (tid < KS * CH) X[tid] = hin[(size_t)g * KS * CH + tid];
    if (tid < CH) {
        float a = bq[tid];
        for (int c = 0; c < CH; ++c) a += S2[c] * Wq[c * CH + tid];
        Qp[tid] = a;
    }
    __syncthreads();
    if (tid < KS * CH) {
        int i = tid / CH, j = tid % CH;
        float ak = bk[j], av = bv[j];
        for (int c = 0; c < CH; ++c) {
            float xv = X[i * CH + c];
            ak += xv * Wk[c * CH + j]; av += xv * Wv[c * CH + j];
        }
        Kp[tid] = ak; Vp[tid] = av;
    }
    __syncthreads();
    if (tid < HEADS) {
        int h = tid;
        const float scale = 0.2357022603955158f;
        float sc[KS], m = -3.0e38f;
        for (int k = 0; k < KS; ++k) {
            float s = 0.f;
            for (int d = 0; d < 3; ++d) s += Qp[h*3 + d] * Kp[k*CH + h*3 + d];
            sc[k] = s * scale; m = fmaxf(m, sc[k]);
        }
        float dsum = 0.f;
        for (int k = 0; k < KS; ++k) { sc[k] = __expf(sc[k] - m); dsum += sc[k]; }
        for (int d = 0; d < 3; ++d) {
            float o = 0.f;
            for (int k = 0; k < KS; ++k) o += sc[k] * Vp[k*CH + h*3 + d];
            O[h*3 + d] = Qp[h*3 + d] + o / dsum;
        }
    }
    __syncthreads();
    if (tid < CH) {
        float t = bo[tid];
        for (int c = 0; c < CH; ++c) t += O[c] * Wo[c * CH + tid];
        pooled[(size_t)g * CH + tid] = O[tid] + fmaxf(t, 0.f);
    }
}

// =====================================================================
// advantage stream per node + per-graph sum (for mean)
// =====================================================================
__global__ void adv_kernel(const float* __restrict__ xc,
                           const float* __restrict__ Wa1, const float* __restrict__ ba1,
                           const float* __restrict__ Wa2, const float* __restrict__ ba2,
                           float* __restrict__ adv, float* __restrict__ madv) {
    int n = blockIdx.x * blockDim.x + threadIdx.x;
    if (n >= NN) return;
    const float* xr = xc + (size_t)n * CH;
    float h[8];
#pragma unroll
    for (int j = 0; j < 8; ++j) h[j] = ba1[j];
    for (int c = 0; c < CH; ++c) {
        float xv = xr[c];
#pragma unroll
        for (int j = 0; j < 8; ++j) h[j] += xv * Wa1[c * 8 + j];
    }
    float a = ba2[0];
#pragma unroll
    for (int j = 0; j < 8; ++j) a += fmaxf(h[j], 0.f) * Wa2[j];
    adv[n] = a;
    atomicAdd(&madv[n >> 10], a);
}

// =====================================================================
// value stream + dueling combine
// =====================================================================
__global__ void final_kernel(const float* __restrict__ xc, const float* __restrict__ pooled,
                             const float* __restrict__ Wv1, const float* __restrict__ bv1,
                             const float* __restrict__ Wv2, const float* __restrict__ bv2,
                             const float* __restrict__ adv, const float* __restrict__ madv,
                             float* __restrict__ outp) {
    int n = blockIdx.x * blockDim.x + threadIdx.x;
    if (n >= NN) return;
    int g = n >> 10;
    const float* pr = pooled + (size_t)g * CH;
    const float* xr = xc + (size_t)n * CH;
    float h[8];
#pragma unroll
    for (int j = 0; j < 8; ++j) h[j] = bv1[j];
    for (int c = 0; c < CH; ++c) {
        float pv = pr[c];
#pragma unroll
        for (int j = 0; j < 8; ++j) h[j] += pv * Wv1[c * 8 + j];
    }
    for (int c = 0; c < CH; ++c) {
        float xv = xr[c];
#pragma unroll
        for (int j = 0; j < 8; ++j) h[j] += xv * Wv1[(CH + c) * 8 + j];
    }
    float val = bv2[0];
#pragma unroll
    for (int j = 0; j < 8; ++j) val += fmaxf(h[j], 0.f) * Wv2[j];
    outp[n] = val + adv[n] - madv[g] * (1.f / NPG);
}

// =====================================================================
extern "C" void kernel_launch(void* const* d_in, const int* in_sizes, int n_in,
                              void* d_out, int out_size, void* d_ws, size_t ws_size,
                              hipStream_t stream) {
    const float* x    = (const float*)d_in[0];
    const int*   ei   = (const int*)d_in[1];
    const int*   srcI = ei;
    const int*   dstI = ei + EE;
    const float* ea   = (const float*)d_in[2];
    // d_in[3] = batch (implied by contiguous graphs)
    const float* P[68];
    for (int i = 4; i < 68 && i < n_in; ++i) P[i] = (const float*)d_in[i];

    // ---- workspace layout (floats) ----
    float* ws    = (float*)d_ws;
    float* xc    = ws;                                  // 18N
    float* ar    = ws + (size_t)18 * NN;                // arena (44N)
    float* xl    = ar;                                  // 8N
    float* xr    = ar + (size_t)8  * NN;                // 8N
    float* cout_ = ar + (size_t)16 * NN;                // 8N
    float* sbuf  = ar + (size_t)24 * NN;                // 13N  (E+N)
    float* den   = ar + (size_t)37 * NN;                // N
    unsigned* smax = (unsigned*)(ar + (size_t)38 * NN); // N
    float* deg   = ar + (size_t)39 * NN;                // N
    float* lattr = ar + (size_t)40 * NN;                // 4N  -> arena end 44N
    float* Kp    = ar;                                  // 18N (aliases conv scratch, post-conv)
    float* Vp    = ar + (size_t)18 * NN;                // 18N
    float* adv   = ws + (size_t)62 * NN;                // N
    float* madv  = ws + (size_t)63 * NN;                // B
    float* qp1   = madv + BB;                           // 144
    float* h1    = qp1 + KS * CH;                       // B*144
    float* h2    = h1 + (size_t)BB * KS * CH;           // B*144
    float* pooled= h2 + (size_t)BB * KS * CH;           // B*18

    const int gridE  = (ET + 255) / 256;
    const int gridEd = (EE + 255) / 256;
    const int gridN  = (NN + 255) / 256;
    const int gemmBlocks = NN / 128;                    // 128 rows / block (exact)

    // ---- self-loop attr (fill_value='mean') ----
    hipMemsetAsync(deg,   0, (size_t)NN * 4, stream);
    hipMemsetAsync(lattr, 0, (size_t)NN * 16, stream);
    hipMemsetAsync(madv,  0, (size_t)BB * 4, stream);
    deg_loop_accum<<<gridEd, 256, 0, stream>>>(dstI, ea, deg, lattr);
    deg_loop_final<<<gridN, 256, 0, stream>>>(lattr, deg);

    // ---- three GATv2 + GraphNorm + ReLU layers ----
    // conv1: 8->8 from x (ld 8); conv2: 8->6 from xc cols 0..7 (ld 18);
    // conv3: 6->4 from xc cols 8..13 (ld 18)
    for (int l = 0; l < 3; ++l) {
        int p  = (l == 0) ? 4 : (l == 1) ? 14 : 24;
        int fo = (l == 0) ? 8 : (l == 1) ? 6 : 4;
        int colOff = (l == 0) ? 0 : (l == 1) ? 8 : 14;
        const float* Wl = P[p], *bl = P[p+1], *Wr = P[p+2], *br = P[p+3];
        const float* We = P[p+4], *att = P[p+5], *cb = P[p+6];
        const float* nw = P[p+7], *nb = P[p+8], *nms = P[p+9];
        if (l == 0) {
            gemm_wmma<8, 8, 8><<<gemmBlocks, 256, 0, stream>>>(x, Wl, bl, xl);
            gemm_wmma<8, 8, 8><<<gemmBlocks, 256, 0, stream>>>(x, Wr, br, xr);
        } else if (l == 1) {
            gemm_wmma<8, 6, 18><<<gemmBlocks, 256, 0, stream>>>(xc + 0, Wl, bl, xl);
            gemm_wmma<8, 6, 18><<<gemmBlocks, 256, 0, stream>>>(xc + 0, Wr, br, xr);
        } else {
            gemm_wmma<6, 4, 18><<<gemmBlocks, 256, 0, stream>>>(xc + 8, Wl, bl, xl);
            gemm_wmma<6, 4, 18><<<gemmBlocks, 256, 0, stream>>>(xc + 8, Wr, br, xr);
        }
        hipMemsetAsync(smax,  0, (size_t)NN * 4, stream);           // 0 == enc(-inf floor)
        hipMemsetAsync(den,   0, (size_t)NN * 4, stream);
        hipMemsetAsync(cout_, 0, (size_t)NN * fo * 4, stream);
        edge_score  <<<gridE, 256, 0, stream>>>(xl, xr, srcI, dstI, ea, lattr, We, att, fo, sbuf, smax);
        edge_exp    <<<gridE, 256, 0, stream>>>(dstI, sbuf, smax, den);
        edge_scatter<<<gridE, 256, 0, stream>>>(xl, srcI, dstI, sbuf, den, fo, cout_);
        graphnorm_relu<<<BB, 256, 0, stream>>>(cout_, cb, nw, nb, nms, fo, xc, colOff);
    }

    // ---- advantage stream (per node) ----
    adv_kernel<<<gridN, 256, 0, stream>>>(xc, P[64], P[65], P[66], P[67], adv, madv);

    // ---- GraphMultisetTransformer ----
    // PMA_8: Qp from S1 (shared), Kp/Vp via f32 WMMA GEMM over all 524K nodes
    small_proj<<<1, 256, 0, stream>>>(P[34], P[35], P[36], qp1, KS);          // S1 @ Wq + bq
    gemm_wmma<18, 18, 18><<<gemmBlocks, 256, 0, stream>>>(xc, P[37], P[38], Kp);
    gemm_wmma<18, 18, 18><<<gemmBlocks, 256, 0, stream>>>(xc, P[39], P[40], Vp);
    pma1_attn<<<BB, 256, 0, stream>>>(Kp, Vp, qp1, P[41], P[42], h1);
    // SAB (8x8)
    sab_kernel<<<BB, 256, 0, stream>>>(h1, P[43], P[44], P[45], P[46], P[47], P[48], P[49], P[50], h2);
    // PMA_1
    pma2_kernel<<<BB, 256, 0, stream>>>(h2, P[51], P[52], P[53], P[54], P[55], P[56], P[57], P[58], P[59], pooled);

    // ---- value stream + dueling combine ----
    final_kernel<<<gridN, 256, 0, stream>>>(xc, pooled, P[60], P[61], P[62], P[63],
                                            adv, madv, (float*)d_out);
}